// DFAProb_23244363006183
// MI455X (gfx1250) — compile-verified
//
#include <hip/hip_runtime.h>

// ---------------------------------------------------------------------------
// DFA fuzzy sum-product update, B = 2M rows x 8 states.
//
// Memory-bound streaming kernel:
//   read  s0 (64MB) + a,b (16MB)          (log_s0 input is UNUSED -> skipped)
//   write log_next (64MB) + next (64MB) + accepted (8MB)
//   total ~216MB @ 23.3 TB/s  => ~9.3us floor.  ~60 flops/row => VALU-light,
//   as long as log() uses the v_log_f32 transcendental (2 ops) and the
//   division uses v_rcp_f32 (clamped inputs => 1-ulp paths are safe).
//
// CDNA5 paths: global_load_async_to_lds_b128 staging of s0 (ASYNCcnt) and
// non-temporal cache hints on all single-use streams.
// ---------------------------------------------------------------------------

typedef float v4f __attribute__((ext_vector_type(4)));

#define DFA_EPS   1e-7f
#define DFA_1MEPS (1.0f - 1e-7f)

#if defined(__gfx1250__)
#define DFA_ASYNC_LDS 1
#else
#define DFA_ASYNC_LDS 0
#endif

__global__ __launch_bounds__(256) void dfa_prob_kernel(
    const float* __restrict__ s0,
    const float* __restrict__ ga,
    const float* __restrict__ gb,
    float* __restrict__ out_log,
    float* __restrict__ out_p,
    float* __restrict__ out_acc,
    int B)
{
    const int r = blockIdx.x * 256 + (int)threadIdx.x;
    if (r >= B) return;

    float s[8];

#if DFA_ASYNC_LDS
    // CDNA5 async global->LDS DMA path: stage this lane's 32B row of s0.
    // INST_OFFSET is added to BOTH the global and LDS addresses (ISA 15.18.3),
    // so one LDS-base VGPR + offset:16 fetches the second half of the row.
    __shared__ v4f tile[256 * 2];          // 8KB, naturally 16B-aligned
    {
        const float* gsrc = s0 + (size_t)r * 8u;
        unsigned lds_addr = (unsigned)(uintptr_t)(&tile[threadIdx.x * 2]);
        asm volatile(
            "global_load_async_to_lds_b128 %0, %1, off\n\t"
            "global_load_async_to_lds_b128 %0, %1, off offset:16"
            :
            : "v"(lds_addr), "v"(gsrc)
            : "memory");
    }
#endif

    // Overlap scalar-guard loads with the async copy.
    const float av = __builtin_nontemporal_load(ga + r);
    const float bv = __builtin_nontemporal_load(gb + r);
    const float ab  = av * bv;        // guard a & b
    const float anb = av - ab;        // guard a & ~b  == a*(1-b)
    const float na  = 1.0f - av;      // guard ~a

#if DFA_ASYNC_LDS
    asm volatile("s_wait_asynccnt 0" ::: "memory");
    {
        const v4f lo = tile[threadIdx.x * 2 + 0];
        const v4f hi = tile[threadIdx.x * 2 + 1];
        s[0]=lo.x; s[1]=lo.y; s[2]=lo.z; s[3]=lo.w;
        s[4]=hi.x; s[5]=hi.y; s[6]=hi.z; s[7]=hi.w;
    }
#else
    {
        const v4f* sv = (const v4f*)(s0 + (size_t)r * 8u);
        v4f lo = __builtin_nontemporal_load(sv + 0);
        v4f hi = __builtin_nontemporal_load(sv + 1);
        s[0]=lo.x; s[1]=lo.y; s[2]=lo.z; s[3]=lo.w;
        s[4]=hi.x; s[5]=hi.y; s[6]=hi.z; s[7]=hi.w;
    }
#endif

    // unnorm[ns] = s0[(ns-1)&7]*ab + s0[(ns-2)&7]*anb + s0[ns]*na
    float u[8];
#pragma unroll
    for (int ns = 0; ns < 8; ++ns) {
        u[ns] = fmaf(s[(ns + 7) & 7], ab,
                fmaf(s[(ns + 6) & 7], anb, s[ns] * na));
    }
    const float denom = ((u[0] + u[1]) + (u[2] + u[3]))
                      + ((u[4] + u[5]) + (u[6] + u[7]));
    const float inv = __builtin_amdgcn_rcpf(denom);   // v_rcp_f32, ~1 ulp

    float p[8], lp[8];
#pragma unroll
    for (int ns = 0; ns < 8; ++ns) {
        float q = u[ns] * inv;
        q = fminf(fmaxf(q, DFA_EPS), DFA_1MEPS);      // clip(eps, 1-eps)
        p[ns]  = q;
        lp[ns] = __logf(q);                           // v_log_f32 + mul(ln2)
    }

    // Streaming 128-bit non-temporal stores.
    v4f pl = {p[0], p[1], p[2], p[3]};
    v4f ph = {p[4], p[5], p[6], p[7]};
    v4f ll = {lp[0], lp[1], lp[2], lp[3]};
    v4f lh = {lp[4], lp[5], lp[6], lp[7]};

    v4f* lrow = (v4f*)(out_log + (size_t)r * 8u);
    v4f* prow = (v4f*)(out_p   + (size_t)r * 8u);
    __builtin_nontemporal_store(ll, lrow + 0);
    __builtin_nontemporal_store(lh, lrow + 1);
    __builtin_nontemporal_store(pl, prow + 0);
    __builtin_nontemporal_store(ph, prow + 1);
    __builtin_nontemporal_store(p[7], out_acc + r);   // accepted = next[:,7]
}

extern "C" void kernel_launch(void* const* d_in, const int* in_sizes, int n_in,
                              void* d_out, int out_size, void* d_ws, size_t ws_size,
                              hipStream_t stream) {
    // Inputs (setup_inputs order): [0]=log_s0 (UNUSED), [1]=s0, [2]=a, [3]=b
    const float* s0 = (const float*)d_in[1];
    const float* a  = (const float*)d_in[2];
    const float* b  = (const float*)d_in[3];
    const int B = in_sizes[2];

    // Outputs concatenated in return order: log_next (B*8) | next (B*8) | acc (B)
    float* out      = (float*)d_out;
    float* out_log  = out;
    float* out_p    = out + (size_t)B * 8u;
    float* out_acc  = out + (size_t)B * 16u;

    const int blocks = (B + 255) / 256;
    dfa_prob_kernel<<<blocks, 256, 0, stream>>>(s0, a, b,
                                                out_log, out_p, out_acc, B);
}